// InfoNCE_14422500180351
// MI455X (gfx1250) — compile-verified
//
#include <hip/hip_runtime.h>
#include <math.h>

#define N      768
#define XDIM   768
#define HID    300
#define HPAD   304
#define W1COLS 1536   // X_DIM + Y_DIM

typedef float v2f __attribute__((ext_vector_type(2)));
typedef float v8f __attribute__((ext_vector_type(8)));

// ---------------------------------------------------------------------------
// Kernel 0: zero the pad-row scratch buffer (read target for h >= 300 lanes).
// ---------------------------------------------------------------------------
__global__ __launch_bounds__(256)
void zero_pad_k(float* __restrict__ zbuf)
{
  const int tid = threadIdx.x;
#pragma unroll
  for (int k = tid; k < XDIM; k += 256) zbuf[k] = 0.f;
}

// ---------------------------------------------------------------------------
// Kernel 1: fp32 WMMA GEMM.
//   z==0:  hxbT[h][m] = sum_k x[m,k]*W1[h,k]        + b1[h]   (transposed out)
//   z==1:  hy  [m][h] = sum_k y[m,k]*W1[h,768+k]              (row-major out)
// Hidden padded to 304; pad lanes point at a zeroed buffer so the K-loop is
// branch-free: pure global_load_b64 + v_wmma_f32_16x16x4_f32.
// ---------------------------------------------------------------------------
__global__ __launch_bounds__(128)
void gemm16x16x4_f32(const float* __restrict__ X, const float* __restrict__ Y,
                     const float* __restrict__ W1, const float* __restrict__ b1,
                     const float* __restrict__ zbuf,
                     float* __restrict__ hxbT, float* __restrict__ hy)
{
  const int lane = threadIdx.x;                  // 0..31
  const int mt   = blockIdx.y * 4 + threadIdx.y; // 0..47  (row tile)
  const int ht   = blockIdx.x;                   // 0..18  (hidden tile)
  const int z    = blockIdx.z;                   // 0: x-part, 1: y-part

  const int row  = mt * 16 + (lane & 15);        // A row (M)
  const int h    = ht * 16 + (lane & 15);        // B row in W1 (N of tile)
  const int kgrp = (lane >> 4) * 2;              // K-pair select per half-wave
  const bool hok = (h < HID);

  const float* A     = z ? Y : X;
  const int    wcol0 = z ? XDIM : 0;

  // Per-lane base pointers, computed once. Out-of-range h reads zeros.
  const float* arow = A + (size_t)row * XDIM + kgrp;
  const float* brow = hok ? (W1 + (size_t)h * W1COLS + wcol0 + kgrp) : (zbuf + kgrp);

  v8f c = {};
#pragma unroll 8
  for (int k = 0; k < XDIM; k += 4) {
    const v2f a = *(const v2f*)(arow + k);
    const v2f b = *(const v2f*)(brow + k);
    // D = A(16x4,f32) * B(4x16,f32) + C(16x16,f32)
    c = __builtin_amdgcn_wmma_f32_16x16x4_f32(false, a, false, b,
                                              (short)0, c, false, false);
  }

  const float bias  = (z == 0 && hok) ? b1[h] : 0.f;
  const int   mbase = mt * 16 + 8 * (lane >> 4);
#pragma unroll
  for (int v = 0; v < 8; ++v) {
    const float val = c[v] + bias;
    const int   m   = mbase + v;
    if (z == 0) hxbT[(size_t)h * N + m]    = val;  // transposed, bias folded
    else        hy  [(size_t)m * HPAD + h] = val;  // row-major
  }
}

__device__ __forceinline__ float softplus_f(float x) {
  return fmaxf(x, 0.f) + log1pf(expf(-fabsf(x)));
}

// ---------------------------------------------------------------------------
// Kernel 2: pairwise relu-weighted contraction + per-row logsumexp + t0.
// Each block handles 4 rows i (reuses every hxbT load for 12 FMAs).
// Each thread owns 3 columns j (768 = 3*256).
// ---------------------------------------------------------------------------
__global__ __launch_bounds__(256)
void pairwise_lse(const float* __restrict__ hxbT, const float* __restrict__ hy,
                  const int* __restrict__ perm, const float* __restrict__ W2,
                  const float* __restrict__ b2v,
                  float* __restrict__ t0, float* __restrict__ lse)
{
  __shared__ float sy[4][HPAD];
  __shared__ float sw[HPAD];
  __shared__ float red[256];

  const int tid = threadIdx.x;
  const int i0  = blockIdx.x * 4;
  const float b2 = b2v[0];

  for (int k = tid; k < HPAD; k += 256)
    sw[k] = (k < HID) ? W2[k] : 0.f;
#pragma unroll
  for (int r = 0; r < 4; ++r) {
    const int pi = perm[i0 + r];
    for (int k = tid; k < HPAD; k += 256)
      sy[r][k] = hy[(size_t)pi * HPAD + k];
  }
  __syncthreads();

  const int j0 = tid, j1 = tid + 256, j2 = tid + 512;
  float acc[4][3];
#pragma unroll
  for (int r = 0; r < 4; ++r) { acc[r][0] = acc[r][1] = acc[r][2] = 0.f; }

  for (int k = 0; k < HPAD; ++k) {
    const float wv = sw[k];
    const float* rowp = hxbT + (size_t)k * N;
    const float x0 = rowp[j0], x1 = rowp[j1], x2 = rowp[j2];
#pragma unroll
    for (int r = 0; r < 4; ++r) {
      const float yv = sy[r][k];
      acc[r][0] = fmaf(fmaxf(x0 + yv, 0.f), wv, acc[r][0]);
      acc[r][1] = fmaf(fmaxf(x1 + yv, 0.f), wv, acc[r][1]);
      acc[r][2] = fmaf(fmaxf(x2 + yv, 0.f), wv, acc[r][2]);
    }
  }

  float s[4][3];
#pragma unroll
  for (int r = 0; r < 4; ++r)
#pragma unroll
    for (int c = 0; c < 3; ++c)
      s[r][c] = softplus_f(acc[r][c] + b2);

  // ---- t0 for the 4 rows (uses unpermuted hy) ----
  for (int r = 0; r < 4; ++r) {
    const int i = i0 + r;
    float tp = 0.f;
    for (int k = tid; k < HID; k += 256)
      tp = fmaf(fmaxf(hxbT[(size_t)k * N + i] + hy[(size_t)i * HPAD + k], 0.f),
                W2[k], tp);
    red[tid] = tp; __syncthreads();
    for (int st = 128; st > 0; st >>= 1) {
      if (tid < st) red[tid] += red[tid + st];
      __syncthreads();
    }
    if (tid == 0) t0[i] = softplus_f(red[0] + b2);
    __syncthreads();
  }

  // ---- per-row logsumexp over j ----
  for (int r = 0; r < 4; ++r) {
    float m = fmaxf(fmaxf(s[r][0], s[r][1]), s[r][2]);
    red[tid] = m; __syncthreads();
    for (int st = 128; st > 0; st >>= 1) {
      if (tid < st) red[tid] = fmaxf(red[tid], red[tid + st]);
      __syncthreads();
    }
    const float M = red[0]; __syncthreads();

    float e = expf(s[r][0] - M) + expf(s[r][1] - M) + expf(s[r][2] - M);
    red[tid] = e; __syncthreads();
    for (int st = 128; st > 0; st >>= 1) {
      if (tid < st) red[tid] += red[tid + st];
      __syncthreads();
    }
    if (tid == 0) lse[i0 + r] = M + logf(red[0]);
    __syncthreads();
  }
}

// ---------------------------------------------------------------------------
// Kernel 3: final scalar reduction.
// ---------------------------------------------------------------------------
__global__ __launch_bounds__(256)
void finalize_k(const float* __restrict__ t0, const float* __restrict__ lse,
                float* __restrict__ out)
{
  __shared__ float r0[256], r1[256];
  const int tid = threadIdx.x;
  float s0 = 0.f, s1 = 0.f;
  for (int i = tid; i < N; i += 256) { s0 += t0[i]; s1 += lse[i]; }
  r0[tid] = s0; r1[tid] = s1; __syncthreads();
  for (int st = 128; st > 0; st >>= 1) {
    if (tid < st) { r0[tid] += r0[tid + st]; r1[tid] += r1[tid + st]; }
    __syncthreads();
  }
  if (tid == 0)
    out[0] = r0[0] / (float)N - (r1[0] / (float)N - logf((float)N));
}

extern "C" void kernel_launch(void* const* d_in, const int* in_sizes, int n_in,
                              void* d_out, int out_size, void* d_ws, size_t ws_size,
                              hipStream_t stream) {
  const float* x    = (const float*)d_in[0];
  const float* y    = (const float*)d_in[1];
  const int*   perm = (const int*)  d_in[2];
  const float* W1   = (const float*)d_in[3];
  const float* b1   = (const float*)d_in[4];
  const float* W2   = (const float*)d_in[5];
  const float* b2   = (const float*)d_in[6];
  float* out = (float*)d_out;

  float* hxbT = (float*)d_ws;              // [HPAD][N]
  float* hy   = hxbT + (size_t)HPAD * N;   // [N][HPAD]
  float* t0   = hy   + (size_t)N * HPAD;   // [N]
  float* lse  = t0 + N;                    // [N]
  float* zbuf = lse + N;                   // [XDIM] zero pad rows

  zero_pad_k<<<1, 256, 0, stream>>>(zbuf);

  dim3 gb(HPAD / 16, N / 16 / 4, 2);       // (19, 12, 2)
  dim3 tb(32, 4, 1);
  gemm16x16x4_f32<<<gb, tb, 0, stream>>>(x, y, W1, b1, zbuf, hxbT, hy);

  pairwise_lse<<<N / 4, 256, 0, stream>>>(hxbT, hy, perm, W2, b2, t0, lse);

  finalize_k<<<1, 256, 0, stream>>>(t0, lse, out);
}